// Conv2d_Downsample_73744588472646
// MI455X (gfx1250) — compile-verified
//
#include <hip/hip_runtime.h>

typedef float v2f __attribute__((ext_vector_type(2)));
typedef float v8f __attribute__((ext_vector_type(8)));

#define KDIM 2304                 // 256 channels * 9 taps
#define SCALE (1.0f / 48.0f)      // 1/sqrt(256*3*3)

#define XS_RS 68                  // raw patch row stride (floats)
#define XS_CS 408                 // raw patch channel stride = 6*68
#define XB_RS 68                  // blurred patch row stride
#define XB_CS 204                 // blurred patch channel stride = 3*68
// LDS element offset of tap t (t = c_loc*9 + kh*3 + kw, t in [0,36))
#define TAPOFF(t) ((((t) / 9) * XB_CS) + ((((t) % 9) / 3) * XB_RS) + ((t) % 3))

// ---- pack & pre-scale weights: w2[(kk/2)*512 + k] is a float2 of K-pair (kk, kk+1)
__global__ __launch_bounds__(256) void prep_weights(const float* __restrict__ w,
                                                    float* __restrict__ w2) {
  int flat = blockIdx.x * 256 + threadIdx.x;   // flat = k*2304 + kk  (== OIHW flat index)
  if (flat >= 512 * KDIM) return;
  int k  = flat / KDIM;
  int kk = flat - k * KDIM;                    // kk = c*9 + kh*3 + kw
  float v = w[flat] * SCALE;
  w2[((kk >> 1) * 512 + k) * 2 + (kk & 1)] = v;
}

// ---- fused blur + 3x3 stride-2 conv as implicit GEMM on WMMA f32 16x16x4
__global__ __launch_bounds__(256) void conv_blur_down(
    const float* __restrict__ x,      // [16,256,128,128]
    const float* __restrict__ w2,     // packed scaled weights
    const float* __restrict__ bias,   // [512]
    const float* __restrict__ blur,   // [4,4]
    float* __restrict__ out)          // [16,512,64,64]
{
  __shared__ float xs[4 * XS_CS];     // raw x patch:    4 ch x 6 x 68
  __shared__ float xbs[4 * XB_CS];    // blurred patch:  4 ch x 3 x 68 (65 used)

  const int tid  = threadIdx.x;
  const int lane = tid & 31;
  const int wave = tid >> 5;
  const int wm   = wave >> 1;         // 0..3  (M wave coord)
  const int wn   = wave & 1;          // 0..1  (N wave coord)
  const int hl   = lane >> 4;         // lane half
  const int idx  = lane & 15;

  const int bx  = blockIdx.x;
  const int ow0 = (bx & 1) * 32;      // pixel tile: 32 contiguous ow
  const int oh  = (bx >> 1) & 63;
  const int n   = bx >> 7;
  const int k0  = blockIdx.y * 128;   // 128 output channels per block

  float Bk[16];
#pragma unroll
  for (int i = 0; i < 16; ++i) Bk[i] = blur[i];

  v8f acc0 = {};
  v8f acc1 = {};

  const int nloc  = wn * 16 + idx;    // pixel within 32-wide tile (B lane = N)
  const int nloc2 = nloc * 2;         // stride-2 into blurred patch
  const int kA0   = k0 + wm * 32 + idx;  // A lane = M (channel), subtile 0
  const int kA1   = kA0 + 16;            // subtile 1
  const v2f* __restrict__ w2v = (const v2f*)w2;

  const int xrow0 = 2 * oh - 2;       // x coords of patch origin (pad=2)
  const int xcol0 = 2 * ow0 - 2;

  for (int stage = 0; stage < 64; ++stage) {
    const int cbase = stage * 4;

    // ---- stage raw x patch into LDS (4 ch x 6 x 68), zero-padded borders ----
    for (int i = tid; i < 4 * XS_CS; i += 256) {
      int cl  = i / XS_CS;
      int rem = i - cl * XS_CS;
      int r   = rem / XS_RS;
      int col = rem - r * XS_RS;
      int xr = xrow0 + r;
      int xc = xcol0 + col;
      float v = 0.0f;
      if ((unsigned)xr < 128u && (unsigned)xc < 128u) {
        const float* p = x + (((size_t)(n * 256 + cbase + cl) * 128 + xr) * 128 + xc);
        v = *p;
        if (cbase + 4 < 256) __builtin_prefetch(p + 4 * 128 * 128, 0, 1);
      }
      xs[i] = v;
    }
    __syncthreads();

    // ---- depthwise 4x4 blur into xbs (4 ch x 3 rows x 65 cols) ----
    for (int i = tid; i < 4 * 195; i += 256) {
      int cl  = i / 195;
      int rem = i - cl * 195;
      int s   = rem / 65;
      int col = rem - s * 65;
      const float* base = &xs[cl * XS_CS + s * XS_RS + col];
      float a = 0.0f;
#pragma unroll
      for (int ii = 0; ii < 4; ++ii)
#pragma unroll
        for (int jj = 0; jj < 4; ++jj)
          a += Bk[ii * 4 + jj] * base[ii * XS_RS + jj];
      xbs[cl * XB_CS + s * XB_RS + col] = a;
    }
    __syncthreads();

    // ---- 9 K-chunks of 4 taps -> V_WMMA_F32_16X16X4_F32 ----
#pragma unroll
    for (int chunk = 0; chunk < 9; ++chunk) {
      const int tA = chunk * 4;
      // lane-half selects K-pair {0,1} vs {2,3} of this chunk (one cndmask each)
      int off0 = hl ? TAPOFF(tA + 2) : TAPOFF(tA + 0);
      int off1 = hl ? TAPOFF(tA + 3) : TAPOFF(tA + 1);
      v2f b;
      b.x = xbs[off0 + nloc2];
      b.y = xbs[off1 + nloc2];
      int pair = stage * 18 + chunk * 2 + hl;
      v2f a0 = w2v[pair * 512 + kA0];
      v2f a1 = w2v[pair * 512 + kA1];
      acc0 = __builtin_amdgcn_wmma_f32_16x16x4_f32(false, a0, false, b,
                                                   (short)0, acc0, false, false);
      acc1 = __builtin_amdgcn_wmma_f32_16x16x4_f32(false, a1, false, b,
                                                   (short)0, acc1, false, false);
    }
    __syncthreads();
  }

  // ---- epilogue: bias + store (D: VGPR v -> M = v + 8*hl, lane idx -> N) ----
  const int ow = ow0 + wn * 16 + idx;
#pragma unroll
  for (int v = 0; v < 8; ++v) {
    int ch0 = k0 + wm * 32 + v + 8 * hl;
    int ch1 = ch0 + 16;
    out[(((size_t)n * 512 + ch0) * 64 + oh) * 64 + ow] = acc0[v] + bias[ch0];
    out[(((size_t)n * 512 + ch1) * 64 + oh) * 64 + ow] = acc1[v] + bias[ch1];
  }
}

extern "C" void kernel_launch(void* const* d_in, const int* in_sizes, int n_in,
                              void* d_out, int out_size, void* d_ws, size_t ws_size,
                              hipStream_t stream) {
  const float* x    = (const float*)d_in[0];
  const float* w    = (const float*)d_in[1];
  const float* bias = (const float*)d_in[2];
  const float* blur = (const float*)d_in[3];
  float* w2  = (float*)d_ws;                 // 512*2304 floats = 4.72 MB scratch
  float* out = (float*)d_out;

  const int totalW = 512 * KDIM;
  prep_weights<<<(totalW + 255) / 256, 256, 0, stream>>>(w, w2);
  // grid.x: 2 ow-tiles * 64 oh * 16 n ; grid.y: 512/128 channel groups
  conv_blur_down<<<dim3(2048, 4), 256, 0, stream>>>(x, w2, bias, blur, out);
}